// OnlineTripletLoss_70368744178174
// MI455X (gfx1250) — compile-verified
//
#include <hip/hip_runtime.h>

#define NROWS 8192
#define DDIM  128
#define NTILES (NROWS / 16)   // 512 column tiles of 16
#define KSTEPS (DDIM / 4)     // 32 k-steps of 4 for f32 WMMA

typedef __attribute__((ext_vector_type(2))) float v2f;
typedef __attribute__((ext_vector_type(8))) float v8f;

// Kernel 1: per-row squared norms; also zero the scalar output (stream-ordered
// before the main kernel, so no race; re-run every call -> deterministic).
__global__ void __launch_bounds__(256) triplet_rownorm_kernel(
    const float* __restrict__ E, float* __restrict__ sq, float* __restrict__ out) {
  const int i = blockIdx.x * blockDim.x + threadIdx.x;  // exactly 8192 threads
  if (i == 0) out[0] = 0.0f;
  const float4* row = (const float4*)(E + (size_t)i * DDIM);
  float s = 0.0f;
#pragma unroll
  for (int k = 0; k < DDIM / 4; ++k) {
    float4 v = row[k];
    s = fmaf(v.x, v.x, s);
    s = fmaf(v.y, v.y, s);
    s = fmaf(v.z, v.z, s);
    s = fmaf(v.w, v.w, s);
  }
  sq[i] = s;
}

// Kernel 2: fused f32-WMMA Gram tiles + masked hardest-pos/neg mining.
// One wave per 16-row strip; 8 waves (256 threads) per block; 64 blocks.
// Two independent WMMA accumulator chains per tile for XDL ILP.
__global__ void __launch_bounds__(256) triplet_mine_kernel(
    const float* __restrict__ E, const int* __restrict__ labels,
    const float* __restrict__ sq, float* __restrict__ out) {
  const int lane  = threadIdx.x & 31;
  const int wave  = threadIdx.x >> 5;
  const int strip = blockIdx.x * 8 + wave;  // 0..511, uniform per wave
  const int ibase = strip * 16;
  const int lhalf = lane >> 4;   // 0 or 1: K split across lane halves
  const int l16   = lane & 15;   // M (for A) / N (for B/C/D)

  // Hoist A fragments for this 16-row strip: 32 k-steps x 2 floats per lane.
  // ISA f32 A layout (16x4): lane%16 = M; VGPR0/1 hold K = k0+2*(lane/16) +0/+1.
  v2f afrag[KSTEPS];
  const float* arow = E + (size_t)(ibase + l16) * DDIM + 2 * lhalf;
#pragma unroll
  for (int ks = 0; ks < KSTEPS; ++ks) afrag[ks] = *(const v2f*)(arow + ks * 4);

  // C/D layout: VGPR v, lane -> row = ibase + v + 8*lhalf, col = jbase + l16.
  int   rlab[8];
  float rsq[8];
#pragma unroll
  for (int v = 0; v < 8; ++v) {
    const int r = ibase + v + 8 * lhalf;
    rlab[v] = labels[r];
    rsq[v]  = sq[r];
  }

  const float INF = __builtin_inff();
  float hmax[8], hmin[8];
#pragma unroll
  for (int v = 0; v < 8; ++v) { hmax[v] = -INF; hmin[v] = INF; }

  for (int jt = 0; jt < NTILES; ++jt) {
    const int   jcol = jt * 16 + l16;
    const int   clab = labels[jcol];
    const float csq  = sq[jcol];
    const float* brow = E + (size_t)jcol * DDIM + 2 * lhalf;  // B = E^T fragment

    // Prefetch next tile's B row into cache (global_prefetch_b8).
    if (jt + 1 < NTILES) {
      __builtin_prefetch(E + (size_t)(jcol + 16) * DDIM + 2 * lhalf, 0, 0);
    }

    // Two independent accumulator chains (even/odd k-steps) -> WMMA ILP.
    v8f acc0 = {0.f, 0.f, 0.f, 0.f, 0.f, 0.f, 0.f, 0.f};
    v8f acc1 = {0.f, 0.f, 0.f, 0.f, 0.f, 0.f, 0.f, 0.f};
#pragma unroll
    for (int ks = 0; ks < KSTEPS; ks += 2) {
      v2f b0 = *(const v2f*)(brow + ks * 4);
      v2f b1 = *(const v2f*)(brow + ks * 4 + 4);
      acc0 = __builtin_amdgcn_wmma_f32_16x16x4_f32(
          false, afrag[ks], false, b0, (short)0, acc0, false, false);
      acc1 = __builtin_amdgcn_wmma_f32_16x16x4_f32(
          false, afrag[ks + 1], false, b1, (short)0, acc1, false, false);
    }

    // Fused epilogue: d2 -> guarded sqrt -> masked running max/min.
#pragma unroll
    for (int v = 0; v < 8; ++v) {
      const float g = acc0[v] + acc1[v];
      float d2 = rsq[v] + csq - 2.0f * g;
      d2 = fmaxf(d2, 0.0f);
      const float dist = (d2 > 0.0f) ? __builtin_sqrtf(d2) : 0.0f;
      const int  row  = ibase + v + 8 * lhalf;
      const bool same = (rlab[v] == clab);
      const bool diag = (row == jcol);
      float p = same ? dist : INF;    // faithful: non-positive pairs get +inf
      float q = same ? -INF : dist;   // positive pairs get -inf in neg min
      if (diag) { p = -INF; q = -INF; }
      hmax[v] = fmaxf(hmax[v], p);
      hmin[v] = fminf(hmin[v], q);
    }
  }

  // Row values live across 16 lanes of each half; xor 1/2/4/8 stays in-half.
#pragma unroll
  for (int v = 0; v < 8; ++v) {
#pragma unroll
    for (int m = 1; m <= 8; m <<= 1) {
      hmax[v] = fmaxf(hmax[v], __shfl_xor(hmax[v], m, 32));
      hmin[v] = fminf(hmin[v], __shfl_xor(hmin[v], m, 32));
    }
  }

  // Lane 0 owns rows ibase..+7, lane 16 owns rows ibase+8..+15.
  if (l16 == 0) {
    float s = 0.0f;
#pragma unroll
    for (int v = 0; v < 8; ++v) s += fmaxf(hmax[v] - hmin[v] + 1.0f, 0.0f);
    atomicAdd(out, s * (1.0f / (float)NROWS));
  }
}

extern "C" void kernel_launch(void* const* d_in, const int* in_sizes, int n_in,
                              void* d_out, int out_size, void* d_ws, size_t ws_size,
                              hipStream_t stream) {
  (void)in_sizes; (void)n_in; (void)out_size; (void)ws_size;
  const float* E      = (const float*)d_in[0];  // [8192,128] f32
  const int*   labels = (const int*)d_in[1];    // [8192] i32
  float*       out    = (float*)d_out;          // scalar f32
  float*       sq     = (float*)d_ws;           // 8192 f32 scratch

  triplet_rownorm_kernel<<<NROWS / 256, 256, 0, stream>>>(E, sq, out);
  triplet_mine_kernel<<<NTILES / 8, 256, 0, stream>>>(E, labels, sq, out);
}